// MDN_83648783057097
// MI455X (gfx1250) — compile-verified
//
#include <hip/hip_runtime.h>
#include <hip/hip_bf16.h>

#define B_   32
#define L_   256
#define T_   1600
#define C_   40
#define K2_  80
#define PADM 1.0e12f

typedef float v2f __attribute__((ext_vector_type(2)));
typedef float v8f __attribute__((ext_vector_type(8)));

__device__ __forceinline__ float logaddexpf_(float x, float y) {
    float mx = fmaxf(x, y);
    float d  = fabsf(x - y);
    return mx + log1pf(expf(-d));
}

__device__ __forceinline__ int modp(int x, int m) {
    int r = x % m;
    return (r < 0) ? r + m : r;
}

// ---------------------------------------------------------------------------
// K1: per-(b,l) preprocessing.
//   Aprep[b,l, 0..39]  = exp(-logvar)            (invvar)
//   Aprep[b,l,40..79]  = -2 * mu * invvar
//   dbias[b,l]         = -0.0125 * sum(mu^2*invvar) - 0.5 * mean(logvar)
// ---------------------------------------------------------------------------
__global__ void mdn_prep(const float* __restrict__ mlv,
                         float* __restrict__ Aprep,
                         float* __restrict__ dbias) {
    int idx = blockIdx.x * blockDim.x + threadIdx.x;      // b*L + l
    if (idx >= B_ * L_) return;
    const float* row  = mlv + (size_t)idx * K2_;          // [mu(40) | logvar(40)]
    float*       arow = Aprep + (size_t)idx * K2_;
    float t3 = 0.f, slv = 0.f;
#pragma unroll 8
    for (int c = 0; c < C_; ++c) {
        float mu = row[c];
        float lv = row[C_ + c];
        float iv = expf(-lv);
        arow[c]        = iv;
        arow[C_ + c]   = -2.f * mu * iv;
        t3  += mu * mu * iv;
        slv += lv;
    }
    dbias[idx] = -0.0125f * t3 - 0.5f * (slv * (1.f / (float)C_));
}

// ---------------------------------------------------------------------------
// K2: lp[b,l,t] = -0.0125 * (Aprep[b,l,:] . Bz[b,:,t]) + dbias[b,l]
//     where Bz rows 0..39 = z^2, rows 40..79 = z.
// fp32 WMMA 16x16x4, block tile 64x64, 8 waves = 4(M) x 2(N), wave = 16x32.
// ---------------------------------------------------------------------------
__global__ __launch_bounds__(256)
void mdn_gemm(const float* __restrict__ Aprep,
              const float* __restrict__ dbias,
              const float* __restrict__ z,
              float* __restrict__ lp) {
    const int b    = blockIdx.z;
    const int Lt   = blockIdx.y * 64;
    const int Tt   = blockIdx.x * 64;
    const int wave = threadIdx.x >> 5;
    const int lane = threadIdx.x & 31;
    const int wm   = wave & 3;                 // 4 waves along M
    const int wn   = wave >> 2;                // 2 waves along N
    const int m    = lane & 15;
    const int hi   = lane >> 4;

    const int row0 = Lt + wm * 16;
    const int col0 = Tt + wn * 32;
    const float* arow = Aprep + ((size_t)(b * L_ + row0 + m)) * K2_;
    const float* zb   = z + (size_t)b * C_ * T_;

    v8f acc0 = {0.f, 0.f, 0.f, 0.f, 0.f, 0.f, 0.f, 0.f};
    v8f acc1 = acc0;

    const int n0 = col0 + m;                   // lane's column in tile 0
    const int n1 = n0 + 16;                    // tile 1

    // K = 0..39 : B rows are z^2
    for (int k0 = 0; k0 < C_; k0 += 4) {
        int ka = k0 + 2 * hi;                  // A/B layout: lanes16-31 hold K+2
        v2f a;  a.x = arow[ka];  a.y = arow[ka + 1];
        float z00 = zb[(size_t)ka * T_ + n0];
        float z01 = zb[(size_t)(ka + 1) * T_ + n0];
        float z10 = zb[(size_t)ka * T_ + n1];
        float z11 = zb[(size_t)(ka + 1) * T_ + n1];
        v2f b0; b0.x = z00 * z00; b0.y = z01 * z01;
        v2f b1; b1.x = z10 * z10; b1.y = z11 * z11;
        acc0 = __builtin_amdgcn_wmma_f32_16x16x4_f32(false, a, false, b0,
                                                     (short)0, acc0, false, false);
        acc1 = __builtin_amdgcn_wmma_f32_16x16x4_f32(false, a, false, b1,
                                                     (short)0, acc1, false, false);
    }
    // K = 40..79 : B rows are z
    for (int k0 = C_; k0 < K2_; k0 += 4) {
        int ka = k0 + 2 * hi;
        v2f a;  a.x = arow[ka];  a.y = arow[ka + 1];
        int kz = ka - C_;
        v2f b0; b0.x = zb[(size_t)kz * T_ + n0];
        b0.y = zb[(size_t)(kz + 1) * T_ + n0];
        v2f b1; b1.x = zb[(size_t)kz * T_ + n1];
        b1.y = zb[(size_t)(kz + 1) * T_ + n1];
        acc0 = __builtin_amdgcn_wmma_f32_16x16x4_f32(false, a, false, b0,
                                                     (short)0, acc0, false, false);
        acc1 = __builtin_amdgcn_wmma_f32_16x16x4_f32(false, a, false, b1,
                                                     (short)0, acc1, false, false);
    }

#pragma unroll
    for (int r = 0; r < 8; ++r) {
        int row = row0 + r + 8 * hi;           // C/D layout: VGPR r -> M = r + 8*hi
        float db = dbias[b * L_ + row];
        size_t base = ((size_t)(b * L_ + row)) * T_;
        lp[base + n0] = -0.0125f * acc0[r] + db;
        lp[base + n1] = -0.0125f * acc1[r] + db;
    }
}

// ---------------------------------------------------------------------------
// K3: forward DP (logaddexp scan). One wave32 per batch, entire alpha[256]
// held in registers (8 per lane). Left-neighbor via one __shfl_up per step:
// no LDS, no barriers, 8-way ILP per lane.
// ---------------------------------------------------------------------------
__global__ __launch_bounds__(32)
void mdn_forward(const float* __restrict__ lp,
                 const int* __restrict__ tlen,
                 const int* __restrict__ mlen,
                 float* __restrict__ loss) {
    const int b  = blockIdx.x;
    const int ln = threadIdx.x;                // lane 0..31
    const int l0 = ln * 8;                     // this lane's l range [l0, l0+8)
    const int ml = mlen[b];
    const int tg = tlen[b] - 1;                // target text row
    const bool mine = ((tg >> 3) == ln);
    const int  tj   = tg & 7;
    const float* lpb = lp + ((size_t)b * L_ + l0) * T_;   // lane-local lp base

    float a[8];
#pragma unroll
    for (int j = 0; j < 8; ++j) a[j] = -PADM;
    if (ln == 0) a[0] = lp[(size_t)b * L_ * T_];          // lp[b,0,0]
    if (ml - 1 == 0 && mine) loss[b] = -(a[tj] / (float)ml);

    for (int t = 1; t < T_; ++t) {
        float left = __shfl_up(a[7], 1);       // previous lane's last alpha
        if (ln == 0) left = -PADM;
        // descending j: a[j-1] still old when computing a[j]
#pragma unroll
        for (int j = 7; j >= 1; --j)
            a[j] = logaddexpf_(a[j] + 1e-7f, a[j - 1] + 1e-7f)
                 + lpb[(size_t)j * T_ + t];
        a[0] = logaddexpf_(a[0] + 1e-7f, left + 1e-7f) + lpb[t];
        if (t == ml - 1 && mine) loss[b] = -(a[tj] / (float)ml);
    }
}

// ---------------------------------------------------------------------------
// K4: Viterbi forward max-scan, same register-resident layout. Writes
// log_beta as [B, T, L] (coalesced float4 stores) into the alignment
// output region, which serves as scratch until K6 overwrites it.
// ---------------------------------------------------------------------------
__global__ __launch_bounds__(32)
void mdn_vitfwd(const float* __restrict__ lp, float* __restrict__ LB) {
    const int b  = blockIdx.x;
    const int ln = threadIdx.x;
    const int l0 = ln * 8;
    const float* lpb = lp + ((size_t)b * L_ + l0) * T_;
    float* lbb = LB + (size_t)b * T_ * L_ + l0;           // [T][L] row base

    float a[8];
#pragma unroll
    for (int j = 0; j < 8; ++j) a[j] = -PADM;
    if (ln == 0) a[0] = lp[(size_t)b * L_ * T_];

    // t = 0 row
    {
        float4 lo = make_float4(a[0], a[1], a[2], a[3]);
        float4 hi = make_float4(a[4], a[5], a[6], a[7]);
        *(float4*)(lbb)     = lo;
        *(float4*)(lbb + 4) = hi;
    }
    for (int t = 1; t < T_; ++t) {
        float left = __shfl_up(a[7], 1);
        if (ln == 0) left = -PADM;
#pragma unroll
        for (int j = 7; j >= 1; --j)
            a[j] = fmaxf(a[j], a[j - 1]) + lpb[(size_t)j * T_ + t];
        a[0] = fmaxf(a[0], left) + lpb[t];
        float4 lo = make_float4(a[0], a[1], a[2], a[3]);
        float4 hi = make_float4(a[4], a[5], a[6], a[7]);
        float* dst = lbb + (size_t)t * L_;
        *(float4*)(dst)     = lo;
        *(float4*)(dst + 4) = hi;
    }
}

// ---------------------------------------------------------------------------
// K5: sequential backtrace, one lane per batch. Mirrors the reference's
// float relu arithmetic and Python-style negative-index wrapping.
// ---------------------------------------------------------------------------
__global__ void mdn_trace(const float* __restrict__ LB,
                          const int* __restrict__ tlen,
                          const int* __restrict__ mlen,
                          int* __restrict__ path) {
    int b = threadIdx.x;
    if (b >= B_) return;
    const float* lbb = LB + (size_t)b * T_ * L_;
    float r = (float)(tlen[b] - 1);
    float c = (float)(mlen[b] - 1);
    path[b * T_ + (T_ - 1)] = (int)r;          // full[0] lands at chrono t = T-1
    for (int k = 1; k < T_; ++k) {
        int ri  = (int)r;
        int ci  = (int)c;
        int cim = modp(ci - 1, T_);
        float vgo = lbb[(size_t)cim * L_ + modp(ri - 1, L_)];
        float vst = lbb[(size_t)cim * L_ + modp(ri,     L_)];
        float is_go = (vgo > vst) ? 1.f : 0.f;
        r = fmaxf(r - is_go + 1.f, 0.f) - 1.f;
        c = fmaxf(c, 0.f) - 1.f;
        path[b * T_ + (T_ - 1 - k)] = (int)r;
    }
}

// ---------------------------------------------------------------------------
// K6: overwrite alignment region with shifted one-hot:
//   align[b,t,l] = (t < ml) && (path[b, t + T - ml] == l)
// ---------------------------------------------------------------------------
__global__ __launch_bounds__(256)
void mdn_align(const int* __restrict__ path,
               const int* __restrict__ mlen,
               float* __restrict__ align) {
    int t = blockIdx.x;
    int b = blockIdx.y;
    int l = threadIdx.x;
    int ml = mlen[b];
    float v = 0.f;
    if (t < ml) {
        int pr = path[b * T_ + (t + T_ - ml)];
        v = (pr == l) ? 1.f : 0.f;
    }
    align[((size_t)b * T_ + t) * L_ + l] = v;
}

extern "C" void kernel_launch(void* const* d_in, const int* in_sizes, int n_in,
                              void* d_out, int out_size, void* d_ws, size_t ws_size,
                              hipStream_t stream) {
    (void)in_sizes; (void)n_in; (void)out_size; (void)ws_size;
    const float* mlv  = (const float*)d_in[0];   // [B, L, 80]
    const float* z    = (const float*)d_in[1];   // [B, 40, T]
    const int*   tlen = (const int*)d_in[2];     // [B]
    const int*   mlen = (const int*)d_in[3];     // [B]

    float* out   = (float*)d_out;
    float* loss  = out;                               // [B]
    float* align = out + B_;                          // [B, T, L]
    float* lp    = align + (size_t)B_ * T_ * L_;      // [B, L, T]

    float* Aprep = (float*)d_ws;                      // [B, L, 80]
    float* dbias = Aprep + (size_t)B_ * L_ * K2_;     // [B, L]
    int*   path  = (int*)(dbias + B_ * L_);           // [B, T]

    mdn_prep<<<(B_ * L_ + 255) / 256, 256, 0, stream>>>(mlv, Aprep, dbias);

    dim3 g2(T_ / 64, L_ / 64, B_);                    // 25 x 4 x 32
    mdn_gemm<<<g2, 256, 0, stream>>>(Aprep, dbias, z, lp);

    mdn_forward<<<B_, 32, 0, stream>>>(lp, tlen, mlen, loss);

    mdn_vitfwd<<<B_, 32, 0, stream>>>(lp, align);     // align region = log_beta scratch

    mdn_trace<<<1, 32, 0, stream>>>(align, tlen, mlen, path);

    dim3 g6(T_, B_);
    mdn_align<<<g6, L_, 0, stream>>>(path, mlen, align);
}